// KnnGNN_51187420233971
// MI455X (gfx1250) — compile-verified
//
#include <hip/hip_runtime.h>

// ---------------------------------------------------------------------------
// KnnGNN on MI455X (gfx1250, wave32).
//
// Roofline: GEMMs total ~38 GFLOP (negligible vs matrix pipes); edge
// gather/scatter moves ~18 GB + ~870M fp32 atomics and dominates at 23.3 TB/s
// (h1/h2/acc are 51 MB each -> resident in the 192 MB L2). GEMMs use
// V_WMMA_F32_16X16X4_F32 to keep full fp32 accuracy. GEMM blocks process two
// 16-row tiles to halve weight traffic; all inner-loop loads are branch-free
// b64 (float2) loads. Edge kernel: 1 wave/edge, float4/lane -> 128B coalesced
// gathers, global_atomic_add_f32 scatter into L2-resident accumulators.
// ---------------------------------------------------------------------------

#define NN   100000
#define EE   1600000
#define ENCF 128
#define EMBF 128
#define PIF  98
#define KCAT (ENCF + PIF)   // 226

typedef __attribute__((ext_vector_type(2))) float v2f;
typedef __attribute__((ext_vector_type(8))) float v8f;

// ---------------- degree / norm kernels ----------------
__global__ __launch_bounds__(256) void k_deg_init(float* __restrict__ d0,
                                                  float* __restrict__ d1) {
  int i = blockIdx.x * 256 + threadIdx.x;
  if (i < NN) { d0[i] = 1.0f; d1[i] = 1.0f; }   // self-loop contributes 1
}

__global__ __launch_bounds__(256) void k_deg_count(const int* __restrict__ ei,
                                                   float* __restrict__ deg) {
  int e = blockIdx.x * 256 + threadIdx.x;
  if (e < EE) atomicAdd(&deg[ei[EE + e]], 1.0f);  // ei[1][e] = target
}

__global__ __launch_bounds__(256) void k_deg_to_dis(float* __restrict__ d0,
                                                    float* __restrict__ d1) {
  int i = blockIdx.x * 256 + threadIdx.x;
  if (i < NN) { d0[i] = rsqrtf(d0[i]); d1[i] = rsqrtf(d1[i]); }  // deg >= 1
}

// ---------------- WMMA helpers ----------------
// f32 16x16x4 VGPR layout (ISA 7.12.2):
//   A (16x4): v0 = K0 (lanes 0-15) / K2 (lanes 16-31); v1 = K1 / K3
//             -> lane holds A[m][k0+2*khalf .. +1], a consecutive float2
//   B (4x16): mirrored, columns striped across lanes (B[k][n] = W[n][k])
//   C/D: VGPR i -> M=i (lanes 0-15), M=8+i (lanes 16-31); N = lane%16
__device__ __forceinline__ v8f wmma4(v2f a, v2f b, v8f c) {
  return __builtin_amdgcn_wmma_f32_16x16x4_f32(false, a, false, b,
                                               (short)0, c, false, false);
}
__device__ __forceinline__ v2f ld2(const float* p) { return *(const v2f*)p; }

// ---------------- dual-weight WMMA GEMM, K = 128 ----------------
// C1 = A @ W1^T, C2 = A @ W2^T.  A: [NN,128], W*: [128,128] row-major.
// Block = 256 threads = 8 waves (one 16-col stripe each); each block does
// TWO 16-row tiles (NN = 3125 * 32) to amortize weight loads.
__global__ __launch_bounds__(256) void k_gemm_dual_128(
    const float* __restrict__ A,
    const float* __restrict__ W1, const float* __restrict__ W2,
    float* __restrict__ C1, float* __restrict__ C2) {
  const int lane  = threadIdx.x & 31;
  const int wave  = threadIdx.x >> 5;
  const int lsub  = lane & 15;
  const int khalf = lane >> 4;
  const int mb    = blockIdx.x * 32;
  const int nb    = wave * 16;
  const long am0  = mb + lsub;          // A row, tile 0
  const long am1  = am0 + 16;           // A row, tile 1
  const long wn   = nb + lsub;          // W row (= output column)

  const float* pA0 = A  + am0 * ENCF + 2 * khalf;
  const float* pA1 = A  + am1 * ENCF + 2 * khalf;
  const float* pW1 = W1 + wn  * ENCF + 2 * khalf;
  const float* pW2 = W2 + wn  * ENCF + 2 * khalf;

  v8f c10 = {}, c20 = {}, c11 = {}, c21 = {};
  for (int k0 = 0; k0 < ENCF; k0 += 4) {
    const v2f w1v = ld2(pW1 + k0);
    const v2f w2v = ld2(pW2 + k0);
    const v2f a0  = ld2(pA0 + k0);
    const v2f a1  = ld2(pA1 + k0);
    c10 = wmma4(a0, w1v, c10);
    c20 = wmma4(a0, w2v, c20);
    c11 = wmma4(a1, w1v, c11);
    c21 = wmma4(a1, w2v, c21);
  }

  const int col = nb + lsub;
#pragma unroll
  for (int i = 0; i < 8; ++i) {
    const long r0 = mb + khalf * 8 + i;
    const long r1 = r0 + 16;
    C1[r0 * EMBF + col] = c10[i];
    C2[r0 * EMBF + col] = c20[i];
    C1[r1 * EMBF + col] = c11[i];
    C2[r1 * EMBF + col] = c21[i];
  }
}

// ---------------- dual-weight WMMA GEMM, K = 226 (concat [x|d2an]) --------
// Branch-free split: K in [0,128) from x, [128,224) from d2an, plus one tail
// chunk (k0=224) where only the khalf==0 half-wave holds data (K=224,225;
// K=226,227 are zero padding).
__global__ __launch_bounds__(256) void k_gemm_dual_cat(
    const float* __restrict__ X, const float* __restrict__ D2,
    const float* __restrict__ W1, const float* __restrict__ W2,
    float* __restrict__ C1, float* __restrict__ C2) {
  const int lane  = threadIdx.x & 31;
  const int wave  = threadIdx.x >> 5;
  const int lsub  = lane & 15;
  const int khalf = lane >> 4;
  const int mb    = blockIdx.x * 32;
  const int nb    = wave * 16;
  const long am0  = mb + lsub;
  const long am1  = am0 + 16;
  const long wn   = nb + lsub;

  const float* pX0 = X  + am0 * ENCF + 2 * khalf;
  const float* pX1 = X  + am1 * ENCF + 2 * khalf;
  const float* pD0 = D2 + am0 * PIF  + 2 * khalf;
  const float* pD1 = D2 + am1 * PIF  + 2 * khalf;
  const float* pW1 = W1 + wn  * KCAT + 2 * khalf;
  const float* pW2 = W2 + wn  * KCAT + 2 * khalf;

  v8f c10 = {}, c20 = {}, c11 = {}, c21 = {};

  // phase 1: K in [0,128) -> x
  for (int k0 = 0; k0 < ENCF; k0 += 4) {
    const v2f w1v = ld2(pW1 + k0);
    const v2f w2v = ld2(pW2 + k0);
    const v2f a0  = ld2(pX0 + k0);
    const v2f a1  = ld2(pX1 + k0);
    c10 = wmma4(a0, w1v, c10);
    c20 = wmma4(a0, w2v, c20);
    c11 = wmma4(a1, w1v, c11);
    c21 = wmma4(a1, w2v, c21);
  }
  // phase 2: K in [128,224) -> d2an[0..96)
  for (int k0 = 0; k0 < 96; k0 += 4) {
    const v2f w1v = ld2(pW1 + ENCF + k0);
    const v2f w2v = ld2(pW2 + ENCF + k0);
    const v2f a0  = ld2(pD0 + k0);
    const v2f a1  = ld2(pD1 + k0);
    c10 = wmma4(a0, w1v, c10);
    c20 = wmma4(a0, w2v, c20);
    c11 = wmma4(a1, w1v, c11);
    c21 = wmma4(a1, w2v, c21);
  }
  // tail: k0 = 224 -> khalf==0 holds K=224,225 (d2an[96,97]); khalf==1 zero
  {
    v2f a0 = {0.0f, 0.0f}, a1 = {0.0f, 0.0f};
    v2f w1v = {0.0f, 0.0f}, w2v = {0.0f, 0.0f};
    if (khalf == 0) {
      a0  = ld2(D2 + am0 * PIF + 96);
      a1  = ld2(D2 + am1 * PIF + 96);
      w1v = ld2(W1 + wn * KCAT + 224);
      w2v = ld2(W2 + wn * KCAT + 224);
    }
    c10 = wmma4(a0, w1v, c10);
    c20 = wmma4(a0, w2v, c20);
    c11 = wmma4(a1, w1v, c11);
    c21 = wmma4(a1, w2v, c21);
  }

  const int col = nb + lsub;
#pragma unroll
  for (int i = 0; i < 8; ++i) {
    const long r0 = mb + khalf * 8 + i;
    const long r1 = r0 + 16;
    C1[r0 * EMBF + col] = c10[i];
    C2[r0 * EMBF + col] = c20[i];
    C1[r1 * EMBF + col] = c11[i];
    C2[r1 * EMBF + col] = c21[i];
  }
}

// ---------------- self-loop init: acc = dis^2 * h1 + 1.0 * h2 --------------
// (self-loop: deg_norm = dis[i]^2, attr = 1 -> einv = 1). Also the
// deterministic re-initialization of acc on every launch. float4 vectorized.
__global__ __launch_bounds__(256) void k_self_init(
    const float* __restrict__ dis, const float* __restrict__ h1,
    const float* __restrict__ h2, float* __restrict__ acc) {
  const long i4 = (long)blockIdx.x * 256 + threadIdx.x;   // float4 index
  if (i4 < (long)NN * (EMBF / 4)) {
    const int node = (int)(i4 >> 5);                      // 32 float4 per row
    const float d = dis[node];
    const float dd = d * d;
    const float4 a = ((const float4*)h1)[i4];
    const float4 b = ((const float4*)h2)[i4];
    float4 r;
    r.x = dd * a.x + b.x; r.y = dd * a.y + b.y;
    r.z = dd * a.z + b.z; r.w = dd * a.w + b.w;
    ((float4*)acc)[i4] = r;
  }
}

// ---------------- edge aggregation (hot loop) ------------------------------
// One wave per edge; lane l handles features [4l, 4l+4).
__global__ __launch_bounds__(256) void k_edge_agg(
    const int* __restrict__ ei, const float* __restrict__ ew,
    const float* __restrict__ dis, const float* __restrict__ h1,
    const float* __restrict__ h2, float* __restrict__ acc) {
  const int lane = threadIdx.x & 31;
  const int wave = threadIdx.x >> 5;
  const long e = (long)blockIdx.x * 8 + wave;
  if (e >= EE) return;

  const int row = ei[e];        // source (x_j)
  const int col = ei[EE + e];   // target (aggregation)
  const float w = ew[e];
  const float dn = dis[row] * dis[col];
  float einv = rsqrtf(w);                       // attr^-0.5
  if (isinf(einv)) einv = 0.0f;                 // attr==0 -> 0
  einv = fminf(einv, 1.0f);                     // clamp >1 to 1

  const float4 a = ((const float4*)(h1 + (long)row * EMBF))[lane];
  const float4 b = ((const float4*)(h2 + (long)row * EMBF))[lane];
  float* dst = acc + (long)col * EMBF + lane * 4;
  atomicAdd(dst + 0, dn * a.x + einv * b.x);
  atomicAdd(dst + 1, dn * a.y + einv * b.y);
  atomicAdd(dst + 2, dn * a.z + einv * b.z);
  atomicAdd(dst + 3, dn * a.w + einv * b.w);
}

// ---------------- 0.5*relu(a0) + 0.5*relu(a1), float4 ----------------------
__global__ __launch_bounds__(256) void k_relu_combine(
    const float* __restrict__ a0, const float* __restrict__ a1,
    float* __restrict__ out) {
  const long i4 = (long)blockIdx.x * 256 + threadIdx.x;
  if (i4 < (long)NN * (EMBF / 4)) {
    const float4 a = ((const float4*)a0)[i4];
    const float4 b = ((const float4*)a1)[i4];
    float4 r;
    r.x = 0.5f * fmaxf(a.x, 0.0f) + 0.5f * fmaxf(b.x, 0.0f);
    r.y = 0.5f * fmaxf(a.y, 0.0f) + 0.5f * fmaxf(b.y, 0.0f);
    r.z = 0.5f * fmaxf(a.z, 0.0f) + 0.5f * fmaxf(b.z, 0.0f);
    r.w = 0.5f * fmaxf(a.w, 0.0f) + 0.5f * fmaxf(b.w, 0.0f);
    ((float4*)out)[i4] = r;
  }
}

// ---------------------------------------------------------------------------
extern "C" void kernel_launch(void* const* d_in, const int* in_sizes, int n_in,
                              void* d_out, int out_size, void* d_ws, size_t ws_size,
                              hipStream_t stream) {
  (void)in_sizes; (void)n_in; (void)out_size; (void)ws_size;

  const float* x      = (const float*)d_in[0];
  const float* d2an   = (const float*)d_in[1];
  const int*   ei0    = (const int*)  d_in[2];
  const float* ew0    = (const float*)d_in[3];
  const int*   ei1    = (const int*)  d_in[4];
  const float* ew1    = (const float*)d_in[5];
  const float* nodeW1 = (const float*)d_in[6];
  const float* nodeW2 = (const float*)d_in[7];
  const float* W1a    = (const float*)d_in[8];
  const float* W1b    = (const float*)d_in[9];
  const float* W2a    = (const float*)d_in[10];
  const float* W2b    = (const float*)d_in[11];
  const float* W3a    = (const float*)d_in[12];
  const float* W3b    = (const float*)d_in[13];
  const float* W4a    = (const float*)d_in[14];
  const float* W4b    = (const float*)d_in[15];
  float* out = (float*)d_out;

  // Workspace carve-up (fp32): 2 small + 6 big buffers (~308 MB total).
  float* ws   = (float*)d_ws;
  float* dis0 = ws;
  float* dis1 = ws + NN;
  const long NB = (long)NN * EMBF;              // 12.8M floats per big buffer
  float* big  = ws + 2 * NN;
  float* xn0  = big + 0 * NB;                   // also reused as xm
  float* xn1  = big + 1 * NB;
  float* h1   = big + 2 * NB;
  float* h2   = big + 3 * NB;
  float* acc0 = big + 4 * NB;
  float* acc1 = big + 5 * NB;
  float* xm   = xn0;                            // xn0 dead after layer 1

  const int gN    = (NN + 255) / 256;           // 391
  const int gE    = (EE + 255) / 256;           // 6250
  const int gEl4  = (int)(NN * (EMBF / 4) / 256);  // 12500 (float4 elems)
  const int gGemm = NN / 32;                    // 3125 (two 16-row tiles/block)
  const int gEdge = EE / 8;                     // 200000 (one wave per edge)

  // degree -> dis (both edge sets)
  k_deg_init  <<<gN, 256, 0, stream>>>(dis0, dis1);
  k_deg_count <<<gE, 256, 0, stream>>>(ei0, dis0);
  k_deg_count <<<gE, 256, 0, stream>>>(ei1, dis1);
  k_deg_to_dis<<<gN, 256, 0, stream>>>(dis0, dis1);

  // node linear: xn0 = [x|d2an] @ nodeW1^T, xn1 = [x|d2an] @ nodeW2^T
  k_gemm_dual_cat<<<gGemm, 256, 0, stream>>>(x, d2an, nodeW1, nodeW2, xn0, xn1);

  // layer 1, edge set 0 -> acc0
  k_gemm_dual_128<<<gGemm, 256, 0, stream>>>(xn0, W1a, W1b, h1, h2);
  k_self_init<<<gEl4, 256, 0, stream>>>(dis0, h1, h2, acc0);
  k_edge_agg <<<gEdge, 256, 0, stream>>>(ei0, ew0, dis0, h1, h2, acc0);

  // layer 1, edge set 1 -> acc1
  k_gemm_dual_128<<<gGemm, 256, 0, stream>>>(xn1, W2a, W2b, h1, h2);
  k_self_init<<<gEl4, 256, 0, stream>>>(dis1, h1, h2, acc1);
  k_edge_agg <<<gEdge, 256, 0, stream>>>(ei1, ew1, dis1, h1, h2, acc1);

  // xm = 0.5*relu(acc0) + 0.5*relu(acc1)
  k_relu_combine<<<gEl4, 256, 0, stream>>>(acc0, acc1, xm);

  // layer 2, edge set 0 -> acc0
  k_gemm_dual_128<<<gGemm, 256, 0, stream>>>(xm, W3a, W3b, h1, h2);
  k_self_init<<<gEl4, 256, 0, stream>>>(dis0, h1, h2, acc0);
  k_edge_agg <<<gEdge, 256, 0, stream>>>(ei0, ew0, dis0, h1, h2, acc0);

  // layer 2, edge set 1 -> acc1
  k_gemm_dual_128<<<gGemm, 256, 0, stream>>>(xm, W4a, W4b, h1, h2);
  k_self_init<<<gEl4, 256, 0, stream>>>(dis1, h1, h2, acc1);
  k_edge_agg <<<gEdge, 256, 0, stream>>>(ei1, ew1, dis1, h1, h2, acc1);

  // final output
  k_relu_combine<<<gEl4, 256, 0, stream>>>(acc0, acc1, out);
}